// BasicDGCNN_88811333747043
// MI455X (gfx1250) — compile-verified
//
#include <hip/hip_runtime.h>
#include <hip/hip_bf16.h>

typedef float v2f __attribute__((ext_vector_type(2)));
typedef float v8f __attribute__((ext_vector_type(8)));

#define EPSBN 1e-5f
#define NPTS  8192
#define BATCH 4
#define KNN   20

__device__ __forceinline__ v8f wmma_f32_4(v2f a, v2f b, v8f c) {
  // D = A(16x4 f32) * B(4x16 f32) + C(16x16 f32)
  return __builtin_amdgcn_wmma_f32_16x16x4_f32(false, a, false, b, (short)0, c,
                                               false, false);
}

// --- CDNA5 async global->LDS copies (ASYNCcnt-tracked DMA path) ------------
__device__ __forceinline__ void async_g2l_b128(const float* g, unsigned lds_off) {
  asm volatile("global_load_async_to_lds_b128 %0, %1, off"
               :: "v"(lds_off), "v"(g) : "memory");
}
__device__ __forceinline__ void async_g2l_b32(const float* g, unsigned lds_off) {
  asm volatile("global_load_async_to_lds_b32 %0, %1, off"
               :: "v"(lds_off), "v"(g) : "memory");
}
__device__ __forceinline__ void wait_async0() {
  asm volatile("s_wait_asynccnt 0x0" ::: "memory");
}
__device__ __forceinline__ unsigned lds_off_of(const void* p) {
  // AS(3) -> generic keeps the workgroup-relative LDS byte offset in addr[31:0]
  return (unsigned)(size_t)p;
}

// ---------------------------------------------------------------------------
// Per-point squared norms
// ---------------------------------------------------------------------------
template <int C>
__global__ void sqnorm_kernel(const float* __restrict__ x, int stride,
                              float* __restrict__ xx) {
  int i = blockIdx.x * blockDim.x + threadIdx.x;
  if (i >= BATCH * NPTS) return;
  const float* p = x + (size_t)i * stride;
  float s = 0.f;
#pragma unroll
  for (int c = 0; c < C; ++c) s += p[c] * p[c];
  xx[i] = s;
}

// ---------------------------------------------------------------------------
// Fused tiled-distance (f32 WMMA) + streaming top-20 selection.
// Block: 128 threads = 4 waves; each wave owns 32 query rows (2 WMMA row
// tiles); each thread owns 1 query for selection. Candidate chunk (64 x C)
// is staged once per block into LDS via async global->LDS DMA, then consumed
// as the WMMA B operand by all 4 waves.  -2 folded into A; ||xj||^2 folded in
// right after accumulation (column is lane-uniform across the 8 acc rows).
// ---------------------------------------------------------------------------
template <int C>
__global__ void __launch_bounds__(128)
knn_kernel(const float* __restrict__ x, int stride,
           const float* __restrict__ xx, int* __restrict__ knn_out) {
  constexpr int CP   = (C + 3) & ~3;
  constexpr int NS   = CP / 4;
  constexpr int XPAD = C + 4;  // row stride: keeps 16B alignment, rotates banks
  __shared__ float ldsD[128][65];    // distance tile (padded rows: no-conflict scan)
  __shared__ float ldsXc[64][XPAD];  // staged candidate features

  const int b    = blockIdx.y;
  const int q0   = blockIdx.x * 128;
  const int tid  = threadIdx.x;
  const int wave = tid >> 5;
  const int lane = tid & 31;
  const int lh   = lane & 15;
  const int koff = (lane < 16) ? 0 : 2;

  const float* xb  = x + (size_t)b * NPTS * stride;
  const float* xxb = xx + (size_t)b * NPTS;

  const int   q   = q0 + tid;
  const float xxq = xxb[q];

  float bd[KNN];
  int   bi[KNN];
#pragma unroll
  for (int t = 0; t < KNN; ++t) { bd[t] = 3.4e38f; bi[t] = 0; }

  // A operand (queries, scaled by -2) is invariant across candidate chunks.
  v2f aA[2][NS];
#pragma unroll
  for (int rt = 0; rt < 2; ++rt) {
    const int qa = q0 + wave * 32 + rt * 16 + lh;
    const float* pq = xb + (size_t)qa * stride;
#pragma unroll
    for (int s = 0; s < NS; ++s) {
      const int c0 = 4 * s + koff;
      aA[rt][s].x = (c0     < C) ? -2.f * pq[c0]     : 0.f;
      aA[rt][s].y = (c0 + 1 < C) ? -2.f * pq[c0 + 1] : 0.f;
    }
  }

  for (int j0 = 0; j0 < NPTS; j0 += 64) {
    // ---- stage 64 candidate rows into LDS with async DMA ----
    if constexpr ((C % 4) == 0) {
      constexpr int V4 = C / 4;
      for (int t = tid; t < 64 * V4; t += 128) {
        const int jj = t / V4, c4 = (t % V4) * 4;
        async_g2l_b128(xb + (size_t)(j0 + jj) * stride + c4,
                       lds_off_of(&ldsXc[jj][c4]));
      }
    } else {
      for (int t = tid; t < 64 * C; t += 128) {
        const int jj = t / C, cc = t % C;
        async_g2l_b32(xb + (size_t)(j0 + jj) * stride + cc,
                      lds_off_of(&ldsXc[jj][cc]));
      }
    }
    wait_async0();
    __syncthreads();  // staging visible; prev scan (ldsD readers) done

    // ---- distance tile: 128 queries x 64 candidates ----
#pragma unroll 1
    for (int ct = 0; ct < 4; ++ct) {
      const int jloc = ct * 16 + lh;
      const float xxj = xxb[j0 + jloc];
      const float* pj = &ldsXc[jloc][0];
      v2f bv[NS];
#pragma unroll
      for (int s = 0; s < NS; ++s) {
        const int c0 = 4 * s + koff;
        bv[s].x = (c0     < C) ? pj[c0]     : 0.f;
        bv[s].y = (c0 + 1 < C) ? pj[c0 + 1] : 0.f;
      }
#pragma unroll
      for (int rt = 0; rt < 2; ++rt) {
        v8f acc = {};
#pragma unroll
        for (int s = 0; s < NS; ++s) acc = wmma_f32_4(aA[rt][s], bv[s], acc);
        const int rbase = wave * 32 + rt * 16 + ((lane < 16) ? 0 : 8);
#pragma unroll
        for (int r = 0; r < 8; ++r)
          ldsD[rbase + r][ct * 16 + lh] = acc[r] + xxj;  // + ||xj||^2 here
      }
    }
    __syncthreads();

    // ---- streaming top-k: register-resident sorted list ----
    const float* drow = ldsD[tid];
#pragma unroll 4
    for (int c = 0; c < 64; ++c) {
      const int j = j0 + c;
      const float d = drow[c] + xxq;
      if (j != q && d < bd[KNN - 1]) {
        float cd = d; int ci = j;
#pragma unroll
        for (int t = 0; t < KNN; ++t) {
          if (cd < bd[t]) {
            float tf = bd[t]; bd[t] = cd; cd = tf;
            int   ti = bi[t]; bi[t] = ci; ci = ti;
          }
        }
      }
    }
    // no barrier needed here: next stage writes only ldsXc, scan reads ldsD;
    // the stage->compute barrier orders ldsD reuse.
  }

  int* op = knn_out + ((size_t)b * NPTS + q) * KNN;
#pragma unroll
  for (int t = 0; t < KNN; ++t) op[t] = bi[t];
}

// ---------------------------------------------------------------------------
// EdgeConv GEMM: rows = edges (4 points * 20 nbrs = 80 rows = 5 tiles,
// 5 waves), K = 2C, cols = O. Edge features built on the fly into the A
// operand; BN + LeakyReLU fused in the accumulator; max over k via LDS.
// ---------------------------------------------------------------------------
template <int C, int O>
__global__ void __launch_bounds__(160)
edge_kernel(const float* __restrict__ xin, int stride,
            const int* __restrict__ knn_idx,
            const float* __restrict__ W,    // [O][2C]
            const float* __restrict__ gma, const float* __restrict__ bta,
            const float* __restrict__ mea, const float* __restrict__ vra,
            float* __restrict__ out, int ostride) {
  constexpr int K2 = 2 * C;
  constexpr int KP = (K2 + 3) & ~3;
  constexpr int NS = KP / 4;
  extern __shared__ float ldsY[];  // [80][O]

  const int b    = blockIdx.y;
  const int n0   = blockIdx.x * 4;
  const int tid  = threadIdx.x;
  const int wave = tid >> 5;
  const int lane = tid & 31;
  const int lh   = lane & 15;
  const int koff = (lane < 16) ? 0 : 2;

  // Edge row this lane supplies in the A operand.
  const int er  = wave * 16 + lh;       // 0..79
  const int p   = er / KNN;             // 0..3
  const int kkE = er % KNN;
  const int n   = n0 + p;
  const int nb  = knn_idx[((size_t)b * NPTS + n) * KNN + kkE];
  const float* xc = xin + ((size_t)b * NPTS + n)  * stride;
  const float* xn = xin + ((size_t)b * NPTS + nb) * stride;

  v2f av[NS];
#pragma unroll
  for (int s = 0; s < NS; ++s) {
    const int c0 = 4 * s + koff;
    const int c1 = c0 + 1;
    av[s].x = (c0 < C) ? xc[c0] : ((c0 < K2) ? (xn[c0 - C] - xc[c0 - C]) : 0.f);
    av[s].y = (c1 < C) ? xc[c1] : ((c1 < K2) ? (xn[c1 - C] - xc[c1 - C]) : 0.f);
  }

#pragma unroll 1
  for (int ctile = 0; ctile < O / 16; ++ctile) {
    const int o = ctile * 16 + lh;
    const float* wr = W + (size_t)o * K2;
    v8f acc = {};
#pragma unroll
    for (int s = 0; s < NS; ++s) {
      const int c0 = 4 * s + koff;
      v2f bw;
      bw.x = (c0     < K2) ? wr[c0]     : 0.f;
      bw.y = (c0 + 1 < K2) ? wr[c0 + 1] : 0.f;
      acc = wmma_f32_4(av[s], bw, acc);
    }
    // Column (channel) is lane-uniform across the 8 accumulator rows.
    const float sc = gma[o] * rsqrtf(vra[o] + EPSBN);
    const float mu = mea[o], bb = bta[o];
    const int rbase = wave * 16 + ((lane < 16) ? 0 : 8);
#pragma unroll
    for (int r = 0; r < 8; ++r) {
      float y = (acc[r] - mu) * sc + bb;
      y = (y >= 0.f) ? y : 0.2f * y;
      ldsY[(size_t)(rbase + r) * O + o] = y;
    }
  }
  __syncthreads();

  for (int t = tid; t < 4 * O; t += 160) {
    const int pp = t / O, oo = t % O;
    float m = -3.4e38f;
#pragma unroll
    for (int kk = 0; kk < KNN; ++kk)
      m = fmaxf(m, ldsY[(size_t)(pp * KNN + kk) * O + oo]);
    out[((size_t)b * NPTS + n0 + pp) * ostride + oo] = m;
  }
}

// ---------------------------------------------------------------------------
// Global conv 224 -> 256 + BN + LeakyReLU + per-block max over 128 points.
// x_concat (29 MB) is L2-resident on MI455X (192 MB L2) so A re-reads per
// column tile are cheap. Deterministic reduction (no float atomics).
// ---------------------------------------------------------------------------
__global__ void __launch_bounds__(128)
global_kernel(const float* __restrict__ xcat,  // [B*N][224]
              const float* __restrict__ Wg,    // [256][224]
              const float* __restrict__ gg, const float* __restrict__ bgp,
              const float* __restrict__ mg, const float* __restrict__ vg,
              float* __restrict__ part) {
  __shared__ float wmax[4][256];
  const int b    = blockIdx.y;
  const int r0   = blockIdx.x * 128;
  const int tid  = threadIdx.x;
  const int wave = tid >> 5;
  const int lane = tid & 31;
  const int lh   = lane & 15;
  const int koff = (lane < 16) ? 0 : 2;
  const float* xb = xcat + (size_t)b * NPTS * 224;

#pragma unroll 1
  for (int ctile = 0; ctile < 16; ++ctile) {
    const int o = ctile * 16 + lh;
    const float* wr = Wg + (size_t)o * 224;
    const float sc = gg[o] * rsqrtf(vg[o] + EPSBN);
    const float mu = mg[o], bb = bgp[o];
    float cm = -3.4e38f;
#pragma unroll 1
    for (int rt = 0; rt < 2; ++rt) {
      const int ra = r0 + wave * 32 + rt * 16 + lh;
      const float* xr = xb + (size_t)ra * 224;
      v8f acc = {};
#pragma unroll 2
      for (int s = 0; s < 56; ++s) {  // K = 224
        const int c0 = 4 * s + koff;
        v2f a, w;
        a.x = xr[c0]; a.y = xr[c0 + 1];
        w.x = wr[c0]; w.y = wr[c0 + 1];
        acc = wmma_f32_4(a, w, acc);
      }
#pragma unroll
      for (int r = 0; r < 8; ++r) {
        float y = (acc[r] - mu) * sc + bb;
        y = (y >= 0.f) ? y : 0.2f * y;
        cm = fmaxf(cm, y);
      }
    }
    cm = fmaxf(cm, __shfl_xor(cm, 16, 32));  // wave32 half-wave share column
    if (lane < 16) wmax[wave][o] = cm;
  }
  __syncthreads();
  for (int t = tid; t < 256; t += 128) {
    float m = fmaxf(fmaxf(wmax[0][t], wmax[1][t]), fmaxf(wmax[2][t], wmax[3][t]));
    part[(((size_t)b * 64) + blockIdx.x) * 256 + t] = m;
  }
}

__global__ void gfeat_reduce_kernel(const float* __restrict__ part,
                                    float* __restrict__ gfeat) {
  const int b = blockIdx.x, o = threadIdx.x;
  float m = -3.4e38f;
  for (int i = 0; i < 64; ++i)
    m = fmaxf(m, part[(((size_t)b * 64) + i) * 256 + o]);
  gfeat[b * 256 + o] = m;
}

// ---------------------------------------------------------------------------
// Vertex / count heads: tiny MLPs, one block per batch element.
// ---------------------------------------------------------------------------
__global__ void __launch_bounds__(256)
heads_kernel(const float* __restrict__ gfeat,
             const float* __restrict__ Wv1, const float* __restrict__ bv1,
             const float* __restrict__ Wv2, const float* __restrict__ bv2,
             const float* __restrict__ Wq1, const float* __restrict__ bq1,
             const float* __restrict__ Wq2, const float* __restrict__ bq2,
             float* __restrict__ vc, float* __restrict__ nv,
             float* __restrict__ nvsp) {
  __shared__ float gf[256];
  __shared__ float h[512];
  __shared__ float qv[64];
  const int b = blockIdx.x, tid = threadIdx.x;
  gf[tid] = gfeat[b * 256 + tid];
  __syncthreads();
  for (int u = tid; u < 512; u += 256) {
    const float* w = Wv1 + (size_t)u * 256;
    float s = bv1[u];
    for (int c = 0; c < 256; ++c) s += w[c] * gf[c];
    h[u] = fmaxf(s, 0.f);
  }
  if (tid < 64) {
    const float* w = Wq1 + (size_t)tid * 256;
    float s = bq1[tid];
    for (int c = 0; c < 256; ++c) s += w[c] * gf[c];
    qv[tid] = fmaxf(s, 0.f);
  }
  __syncthreads();
  if (tid < 192) {
    const float* w = Wv2 + (size_t)tid * 512;
    float s = bv2[tid];
    for (int c = 0; c < 512; ++c) s += w[c] * h[c];
    vc[b * 192 + tid] = s;
  }
  if (tid == 0) {
    float s = bq2[0];
    for (int c = 0; c < 64; ++c) s += Wq2[c] * qv[c];
    const float pr = 1.f / (1.f + __expf(-s));
    nvsp[b] = pr;
    float r = rintf(pr * 64.f);  // round-half-even, matches jnp.round
    r = fminf(fmaxf(r, 1.f), 64.f);
    nv[b] = r;
  }
}

// ---------------------------------------------------------------------------
extern "C" void kernel_launch(void* const* d_in, const int* in_sizes, int n_in,
                              void* d_out, int out_size, void* d_ws,
                              size_t ws_size, hipStream_t stream) {
  (void)in_sizes; (void)n_in; (void)out_size; (void)ws_size;
  const float* x   = (const float*)d_in[0];
  const float* W1  = (const float*)d_in[1];
  const float* g1  = (const float*)d_in[2];
  const float* b1  = (const float*)d_in[3];
  const float* m1  = (const float*)d_in[4];
  const float* v1  = (const float*)d_in[5];
  const float* W2  = (const float*)d_in[6];
  const float* g2  = (const float*)d_in[7];
  const float* b2  = (const float*)d_in[8];
  const float* m2  = (const float*)d_in[9];
  const float* v2  = (const float*)d_in[10];
  const float* W3  = (const float*)d_in[11];
  const float* g3  = (const float*)d_in[12];
  const float* b3  = (const float*)d_in[13];
  const float* m3  = (const float*)d_in[14];
  const float* v3  = (const float*)d_in[15];
  const float* Wg  = (const float*)d_in[16];
  const float* gg  = (const float*)d_in[17];
  const float* bg  = (const float*)d_in[18];
  const float* mg  = (const float*)d_in[19];
  const float* vg  = (const float*)d_in[20];
  const float* Wv1 = (const float*)d_in[21];
  const float* bv1 = (const float*)d_in[22];
  const float* Wv2 = (const float*)d_in[23];
  const float* bv2 = (const float*)d_in[24];
  const float* Wq1 = (const float*)d_in[25];
  const float* bq1 = (const float*)d_in[26];
  const float* Wq2 = (const float*)d_in[27];
  const float* bq2 = (const float*)d_in[28];

  float* out   = (float*)d_out;
  float* vc    = out;          // [4,64,3]  = 768
  float* nv    = out + 768;    // [4]
  float* nvsp  = out + 772;    // [4]
  float* gfeat = out + 776;    // [4,256]   = 1024
  float* xcat  = out + 1800;   // [4,8192,224]

  char*  ws   = (char*)d_ws;
  float* xx   = (float*)ws;                                      // 32768 f32
  int*   idx  = (int*)(ws + (size_t)32768 * 4);                  // 655360 i32
  float* part = (float*)(ws + (size_t)32768 * 4 +
                         (size_t)BATCH * NPTS * KNN * 4);        // 65536 f32

  const int  sqBlocks = (BATCH * NPTS + 255) / 256;
  const dim3 gridQ(NPTS / 128, BATCH);
  const dim3 gridE(NPTS / 4, BATCH);
  const dim3 gridG(NPTS / 128, BATCH);

  // EdgeConv 1: x (C=5, stride 5) -> x1 (32 ch @ xcat+0, stride 224)
  sqnorm_kernel<5><<<sqBlocks, 256, 0, stream>>>(x, 5, xx);
  knn_kernel<5><<<gridQ, 128, 0, stream>>>(x, 5, xx, idx);
  edge_kernel<5, 32><<<gridE, 160, 80 * 32 * 4, stream>>>(
      x, 5, idx, W1, g1, b1, m1, v1, xcat + 0, 224);

  // EdgeConv 2: x1 -> x2 (64 ch @ xcat+32)
  sqnorm_kernel<32><<<sqBlocks, 256, 0, stream>>>(xcat + 0, 224, xx);
  knn_kernel<32><<<gridQ, 128, 0, stream>>>(xcat + 0, 224, xx, idx);
  edge_kernel<32, 64><<<gridE, 160, 80 * 64 * 4, stream>>>(
      xcat + 0, 224, idx, W2, g2, b2, m2, v2, xcat + 32, 224);

  // EdgeConv 3: x2 -> x3 (128 ch @ xcat+96)
  sqnorm_kernel<64><<<sqBlocks, 256, 0, stream>>>(xcat + 32, 224, xx);
  knn_kernel<64><<<gridQ, 128, 0, stream>>>(xcat + 32, 224, xx, idx);
  edge_kernel<64, 128><<<gridE, 160, 80 * 128 * 4, stream>>>(
      xcat + 32, 224, idx, W3, g3, b3, m3, v3, xcat + 96, 224);

  // Global conv + max-pool over points, then heads.
  global_kernel<<<gridG, 128, 0, stream>>>(xcat, Wg, gg, bg, mg, vg, part);
  gfeat_reduce_kernel<<<BATCH, 256, 0, stream>>>(part, gfeat);
  heads_kernel<<<BATCH, 256, 0, stream>>>(gfeat, Wv1, bv1, Wv2, bv2,
                                          Wq1, bq1, Wq2, bq2, vc, nv, nvsp);
}